// i_k1_calc_impl3_50405736186087
// MI455X (gfx1250) — compile-verified
//
#include <hip/hip_runtime.h>
#include <hip/hip_bf16.h>

// ---------------------------------------------------------------------------
// Per-sample physics step driven by a tiny MLP (3->12->8->2, tanh) + 2x2 solve.
// weights^T live in the WMMA A operand (M = output dim), 16 samples per wave in
// the B operand's N dimension. Layer outputs appear in C layout (M across
// VGPRs/lane-halves, N = sample across lanes) and are re-packed into the next
// layer's B operand with lane-half swaps (v_permlanex16 if available, else
// ds_swizzle SWAPX16). One forward pass plus two JVP tangent passes share the
// A operands: 18 x V_WMMA_F32_16X16X4_F32 per 16 samples.
// tanh via hardware v_tanh_f32 when available, else 1 - 2*v_rcp(v_exp(2x)+1).
// ---------------------------------------------------------------------------

typedef float v2f __attribute__((ext_vector_type(2)));
typedef float v8f __attribute__((ext_vector_type(8)));

#define R_S_C   0.18f
#define DT_C    5e-05f
#define W_EL_C  942.4777960769379f   /* 2*3000*pi/60*3 */
#define S1_C    186.66666666666666f  /* U_DC/2 * 2/3    */

static __device__ __forceinline__ v8f wmma4(v2f a, v2f b, v8f c) {
    // D = A(16x4 f32) x B(4x16 f32) + C(16x16 f32)
    return __builtin_amdgcn_wmma_f32_16x16x4_f32(
        /*neg_a=*/false, a, /*neg_b=*/false, b,
        /*c_mod=*/(short)0, c, /*reuse_a=*/false, /*reuse_b=*/false);
}

// Swap the two 16-lane halves of the wave: lane l <-> lane l^16.
static __device__ __forceinline__ float swap16(float v) {
#if __has_builtin(__builtin_amdgcn_permlanex16)
    // VALU cross-half permute with identity lane selects: no DS pipe, no dscnt.
    int i = __builtin_amdgcn_permlanex16(__float_as_int(v), __float_as_int(v),
                                         0x76543210, 0xfedcba98,
                                         /*fi=*/false, /*bound_ctrl=*/false);
    return __int_as_float(i);
#else
    return __int_as_float(__builtin_amdgcn_ds_swizzle(__float_as_int(v), 0x401F));
#endif
}

static __device__ __forceinline__ float fast_tanh(float x) {
#if __has_builtin(__builtin_amdgcn_tanhf)
    return __builtin_amdgcn_tanhf(x);              // gfx1250 v_tanh_f32
#else
    // tanh(x) = 1 - 2/(exp(2x)+1); robust at +/-inf
    float e = __expf(2.0f * x);                    // v_exp_f32 (scale folded)
    float r = __builtin_amdgcn_rcpf(e + 1.0f);     // raw v_rcp_f32
    return __builtin_fmaf(-2.0f, r, 1.0f);
#endif
}

static __device__ __forceinline__ v8f tanh8(v8f z) {
    v8f h;
#pragma unroll
    for (int i = 0; i < 8; ++i) h[i] = fast_tanh(z[i]);
    return h;
}

// result: lower half = x's lower half (in place), upper half = y's lower half.
static __device__ __forceinline__ float mergeLL(float x, float y, bool up) {
    float ys = swap16(y);
    return up ? ys : x;
}
// result: lower half = x's upper half, upper half = y's upper half (in place).
static __device__ __forceinline__ float mergeUU(float x, float y, bool up) {
    float xs = swap16(x);
    return up ? y : xs;
}

// 12->8 layer: h (C layout, rows 0..11 used) times A2[3] (W2^T slices), K=12.
static __device__ __forceinline__ v8f chain3(const v2f A[3], const v8f h, v8f c, bool up) {
    v2f b0, b1, b2;
    b0.x = mergeLL(h[0], h[2], up);  b0.y = mergeLL(h[1], h[3], up);  // K=0..3
    b1.x = mergeLL(h[4], h[6], up);  b1.y = mergeLL(h[5], h[7], up);  // K=4..7
    b2.x = mergeUU(h[0], h[2], up);  b2.y = mergeUU(h[1], h[3], up);  // K=8..11
    c = wmma4(A[0], b0, c);
    c = wmma4(A[1], b1, c);
    c = wmma4(A[2], b2, c);
    return c;
}

// 8->2 layer: g (C layout, rows 0..7 used) times A3[2] (W3^T slices), K=8.
static __device__ __forceinline__ v8f chain2(const v2f A[2], const v8f g, v8f c, bool up) {
    v2f b0, b1;
    b0.x = mergeLL(g[0], g[2], up);  b0.y = mergeLL(g[1], g[3], up);  // K=0..3
    b1.x = mergeLL(g[4], g[6], up);  b1.y = mergeLL(g[5], g[7], up);  // K=4..7
    c = wmma4(A[0], b0, c);
    c = wmma4(A[1], b1, c);
    return c;
}

__global__ __launch_bounds__(256)
void psi_mlp_solver(const float* __restrict__ in,
                    const float* __restrict__ W1, const float* __restrict__ b1,
                    const float* __restrict__ W2, const float* __restrict__ b2,
                    const float* __restrict__ W3, const float* __restrict__ b3,
                    float* __restrict__ out, int B)
{
    const int  lane = (int)(threadIdx.x & 31);
    const int  m    = lane & 15;               // A-operand row (output dim / sample idx)
    const bool up   = (lane & 16) != 0;        // upper lane half
    const int  k0   = up ? 2 : 0;              // K pair handled by this half

    // ---- A operands: transposed weights in WMMA A layout (loop invariant) ----
    v2f A1;
    A1.x = (m < 12)               ? W1[(size_t)k0 * 12 + m]       : 0.0f; // K=k0   (<3)
    A1.y = (m < 12 && k0 + 1 < 3) ? W1[(size_t)(k0 + 1) * 12 + m] : 0.0f; // K=k0+1
    v2f A2[3], A3[2];
#pragma unroll
    for (int j = 0; j < 3; ++j) {
        int kk = 4 * j + k0;
        A2[j].x = (m < 8 && kk     < 12) ? W2[(size_t)kk * 8 + m]       : 0.0f;
        A2[j].y = (m < 8 && kk + 1 < 12) ? W2[(size_t)(kk + 1) * 8 + m] : 0.0f;
    }
#pragma unroll
    for (int j = 0; j < 2; ++j) {
        int kk = 4 * j + k0;
        A3[j].x = (m < 2 && kk     < 8) ? W3[(size_t)kk * 2 + m]       : 0.0f;
        A3[j].y = (m < 2 && kk + 1 < 8) ? W3[(size_t)(kk + 1) * 2 + m] : 0.0f;
    }

    // ---- bias-initialized / zero C accumulators (C layout: row M = v + up*8) ----
    v8f C1b, C2b, C3b, Z;
#pragma unroll
    for (int v = 0; v < 8; ++v) {
        int M  = v + (up ? 8 : 0);
        C1b[v] = (M < 12) ? b1[M] : 0.0f;
        C2b[v] = (M < 8)  ? b2[M] : 0.0f;
        C3b[v] = (M < 2)  ? b3[M] : 0.0f;
        Z[v]   = 0.0f;
    }

    const int ntiles = (B + 15) >> 4;
    const int wid    = (int)((blockIdx.x * blockDim.x + threadIdx.x) >> 5);
    const int nw     = (int)((gridDim.x * blockDim.x) >> 5);

    for (int t = wid; t < ntiles; t += nw) {
        const int  s     = t * 16 + m;
        const bool valid = s < B;
        const int  si    = valid ? s : (B - 1);
        const float* row = in + (size_t)si * 5;
        const float x0 = row[0], x1 = row[1], x2 = row[2];

        // norm and its partials (JVP of ||i||): d norm / d x_j = x_j / norm
        const float nsq   = x0 * x0 + x1 * x1;
        const float inv_n = __frsqrt_rn(nsq);            // raw v_rsq_f32
        const float nrm   = nsq * inv_n;
        const float q0    = x0 * inv_n;
        const float q1    = x1 * inv_n;
        const float nrm_u = swap16(nrm);
        const float q0_u  = swap16(q0);
        const float q1_u  = swap16(q1);

        // B operands for layer 1 (4x16, K = {x0, x1, norm, 0} / tangent rows)
        v2f Bf, Bt1, Bt2;
        Bf.x  = up ? nrm_u : x0;   Bf.y  = up ? 0.0f : x1;
        Bt1.x = up ? q0_u  : 1.0f; Bt1.y = 0.0f;
        Bt2.x = up ? q1_u  : 0.0f; Bt2.y = up ? 0.0f : 1.0f;

        // ---------------- forward pass ----------------
        v8f z1 = wmma4(A1, Bf, C1b);
        v8f h1 = tanh8(z1);
        v8f s1, s2;
#pragma unroll
        for (int i = 0; i < 8; ++i) s1[i] = 1.0f - h1[i] * h1[i];
        v8f z2 = chain3(A2, h1, C2b, up);
        v8f h2 = tanh8(z2);
#pragma unroll
        for (int i = 0; i < 8; ++i) s2[i] = 1.0f - h2[i] * h2[i];
        v8f psi = chain2(A3, h2, C3b, up);

        // ---------------- tangent pass 1 (e1) ----------------
        v8f u1 = wmma4(A1, Bt1, Z);
#pragma unroll
        for (int i = 0; i < 8; ++i) u1[i] *= s1[i];
        v8f u2 = chain3(A2, u1, Z, up);
#pragma unroll
        for (int i = 0; i < 8; ++i) u2[i] *= s2[i];
        v8f d1v = chain2(A3, u2, Z, up);

        // ---------------- tangent pass 2 (e2) ----------------
        v8f w1t = wmma4(A1, Bt2, Z);
#pragma unroll
        for (int i = 0; i < 8; ++i) w1t[i] *= s1[i];
        v8f w2t = chain3(A2, w1t, Z, up);
#pragma unroll
        for (int i = 0; i < 8; ++i) w2t[i] *= s2[i];
        v8f d2v = chain2(A3, w2t, Z, up);

        // ---------------- per-sample 2x2 solve + Euler step (lanes 0-15) ----------------
        const float psi_d = psi[0], psi_q = psi[1];
        const float a  = d1v[0];   // d psi_d / d x0
        const float cc = d1v[1];   // d psi_q / d x0
        const float bb = d2v[0];   // d psi_d / d x1
        const float dd = d2v[1];   // d psi_q / d x1
        const float det    = a * dd - bb * cc;
        const float invdet = __builtin_amdgcn_rcpf(det); // raw v_rcp_f32
        const float rr = -R_S_C * (x0 + x1);
        const float r0 = rr + W_EL_C * psi_q;
        const float r1 = rr + S1_C * x2 - W_EL_C * psi_d;
        const float ik0 = (dd * r0 - bb * r1) * invdet;
        const float ik1 = (a  * r1 - cc * r0) * invdet;

        if (!up && valid) {
            float2 o;
            o.x = ik0 * DT_C + x0;
            o.y = ik1 * DT_C + x1;
            *(float2*)(out + (size_t)s * 2) = o;
        }
    }
}

extern "C" void kernel_launch(void* const* d_in, const int* in_sizes, int n_in,
                              void* d_out, int out_size, void* d_ws, size_t ws_size,
                              hipStream_t stream) {
    (void)n_in; (void)out_size; (void)d_ws; (void)ws_size;
    const float* in = (const float*)d_in[0];
    const float* W1 = (const float*)d_in[1];
    const float* b1 = (const float*)d_in[2];
    const float* W2 = (const float*)d_in[3];
    const float* b2 = (const float*)d_in[4];
    const float* W3 = (const float*)d_in[5];
    const float* b3 = (const float*)d_in[6];
    float* out = (float*)d_out;
    const int B = in_sizes[0] / 5;

    dim3 block(256);           // 8 wave32 per block
    dim3 grid(2048);           // grid-stride over 16-sample tiles
    psi_mlp_solver<<<grid, block, 0, stream>>>(in, W1, b1, W2, b2, W3, b3, out, B);
}